// GCMCGraphConv_6322191859867
// MI455X (gfx1250) — compile-verified
//
#include <hip/hip_runtime.h>
#include <hip/hip_bf16.h>

// CDNA5 / gfx1250: wave32, WMMA f32 16x16x4, async global->LDS copies.
typedef float v2f __attribute__((ext_vector_type(2)));
typedef float v8f __attribute__((ext_vector_type(8)));
typedef int   v4i __attribute__((ext_vector_type(4)));

typedef __attribute__((address_space(1))) v4i* gas_v4i_ptr;
typedef __attribute__((address_space(3))) v4i* las_v4i_ptr;

#define N_NODES 100000
#define N_EDGES 6400000
#define HDIM 16
#define RFDIM 64

// ---- async global->LDS (ASYNCcnt-tracked) ---------------------------------
// Probe-discovered prototype: param0 = v4i addrspace(1)* (non-const).
#if __has_builtin(__builtin_amdgcn_global_load_async_to_lds_b128)
#define ASYNC_COPY16(gptr, lptr)                                            \
  __builtin_amdgcn_global_load_async_to_lds_b128(                           \
      (gas_v4i_ptr)(v4i*)(const_cast<float*>(gptr)),                        \
      (las_v4i_ptr)(v4i*)(lptr), 0, 0)
#else
#define ASYNC_COPY16(gptr, lptr)                                            \
  asm volatile("global_load_async_to_lds_b128 %0, %1, off" ::               \
                   "v"((unsigned)(size_t)(lptr)),                           \
                   "v"((unsigned long long)(size_t)(gptr))                  \
               : "memory")
#endif

#if __has_builtin(__builtin_amdgcn_s_wait_asynccnt)
#define WAIT_ASYNC(n) __builtin_amdgcn_s_wait_asynccnt(n)
#else
#define WAIT_ASYNC(n) asm volatile("s_wait_asynccnt %0" ::"i"(n) : "memory")
#endif

// ---------------------------------------------------------------------------
// Zero the output accumulator (harness poisons d_out with 0xAA).
// ---------------------------------------------------------------------------
__global__ void zero_out_kernel(float* __restrict__ out, int n) {
  int i = blockIdx.x * blockDim.x + threadIdx.x;
  if (i < n) out[i] = 0.0f;
}

// ---------------------------------------------------------------------------
// Kernel 1: h = feat @ W_node^T   [N_NODES,16] = [N_NODES,16] x [16,16]^T
// One wave per 16-node tile, 4 x V_WMMA_F32_16X16X4_F32 (K=16).
// Tiny kernel (6.4 MB total) -> direct WMMA-layout loads are fine.
// ---------------------------------------------------------------------------
__global__ __launch_bounds__(256) void node_h_kernel(
    const float* __restrict__ feat, const float* __restrict__ Wn,
    float* __restrict__ h) {
  const int lane  = threadIdx.x & 31;
  const int row   = lane & 15;   // M for A-loads, N for B-loads / D-stores
  const int khalf = lane >> 4;   // 0: K pair {0,1}, 1: K pair {2,3}

  v2f b[4];
#pragma unroll
  for (int kk = 0; kk < 4; ++kk)
    b[kk] = *(const v2f*)(Wn + row * HDIM + 4 * kk + 2 * khalf);

  const int wave   = (blockIdx.x * blockDim.x + threadIdx.x) >> 5;
  const int nwaves = (gridDim.x * blockDim.x) >> 5;
  const int ntiles = N_NODES / 16;

  for (int t = wave; t < ntiles; t += nwaves) {
    const float* arow = feat + (size_t)(t * 16 + row) * HDIM;
    v2f a[4];
#pragma unroll
    for (int kk = 0; kk < 4; ++kk)
      a[kk] = *(const v2f*)(arow + 4 * kk + 2 * khalf);

    v8f c = {};
#pragma unroll
    for (int kk = 0; kk < 4; ++kk)
      c = __builtin_amdgcn_wmma_f32_16x16x4_f32(
          false, a[kk], false, b[kk], (short)0, c, false, false);

#pragma unroll
    for (int v = 0; v < 8; ++v) {
      const int m = v + 8 * khalf;  // D: vgpr v, lane -> (m, n=lane&15)
      h[(size_t)(t * 16 + m) * HDIM + row] = c[v];
    }
  }
}

// ---------------------------------------------------------------------------
// Kernel 2: per 16-edge tile per wave:
//   - async-copy the contiguous 4KB review_feat tile into LDS (double
//     buffered, XOR-swizzled 16B chunks for conflict-free ds_load_b64)
//   - rf_tile[16x16] = tile[16x64] @ W_review^T via 16x WMMA f32 16x16x4
//   - epilogue: atomicAdd(out[dst], (rf + h[src]) * edge_w)  == segment_sum
// LDS: 8 waves/block * 2 buffers * 4KB = 64KB.
// ---------------------------------------------------------------------------
__global__ __launch_bounds__(256) void edge_kernel(
    const float* __restrict__ review_feat, const float* __restrict__ edge_w,
    const int* __restrict__ src_idx, const int* __restrict__ dst_idx,
    const float* __restrict__ Wr, const float* __restrict__ h,
    float* __restrict__ out) {
  __shared__ __align__(16) float smem[8 * 2 * 1024];  // 64 KB

  const int lane  = threadIdx.x & 31;
  const int row   = lane & 15;
  const int khalf = lane >> 4;
  float* wbuf = smem + (threadIdx.x >> 5) * 2048;  // this wave's two buffers

  // Preload B = W_review^T for all 16 K-chunks (32 VGPRs), reused every tile.
  // B (4x16 f32): lane L -> col N=L%16, K = 4*kk + 2*(L/16)+{0,1};
  // B[k][n] = W_review[n][k] (row-major [16,64]).
  v2f b[16];
#pragma unroll
  for (int kk = 0; kk < 16; ++kk)
    b[kk] = *(const v2f*)(Wr + row * RFDIM + 4 * kk + 2 * khalf);

  const int wave   = (blockIdx.x * blockDim.x + threadIdx.x) >> 5;
  const int nwaves = (gridDim.x * blockDim.x) >> 5;
  const int ntiles = N_EDGES / 16;

  // Issue one tile's 4KB copy: 8 x b128, fully coalesced (512B/instruction).
  // 16B chunk (row r, col c) lands at LDS slot (c ^ r) within row r.
  auto issue_tile = [&](int tile, int d) {
    const float* gbase = review_feat + (size_t)tile * (16 * RFDIM);
    float* lbase = wbuf + d * 1024;
#pragma unroll
    for (int i = 0; i < 8; ++i) {
      const int chunk = i * 32 + lane;  // 16B-chunk index in tile (0..255)
      const int r = chunk >> 4;
      const int c = chunk & 15;
      const int sw = c ^ r;  // XOR swizzle
      ASYNC_COPY16(gbase + chunk * 4, lbase + r * 64 + sw * 4);
    }
  };

  int t = wave;
  int dbuf = 0;
  if (t < ntiles) issue_tile(t, 0);

  for (; t < ntiles; t += nwaves) {
    const int tn = t + nwaves;
    if (tn < ntiles) {
      issue_tile(tn, dbuf ^ 1);
      WAIT_ASYNC(8);  // wait for current tile; next tile stays in flight
    } else {
      WAIT_ASYNC(0);
    }
    asm volatile("" ::: "memory");

    // A (16x4 f32) chunk kk: lane -> row M=L%16, dwords 4kk+2khalf+{0,1}.
    // Chunk kk of row r sits at swizzled slot (kk ^ r): conflict-free b64.
    const float* rbuf = wbuf + dbuf * 1024;
    v2f a[16];
#pragma unroll
    for (int kk = 0; kk < 16; ++kk) {
      const int sw = (kk ^ row) & 15;
      a[kk] = *(const v2f*)(rbuf + row * 64 + sw * 4 + 2 * khalf);
    }

    v8f c = {};
#pragma unroll
    for (int kk = 0; kk < 16; ++kk)
      c = __builtin_amdgcn_wmma_f32_16x16x4_f32(
          false, a[kk], false, b[kk], (short)0, c, false, false);

    // Epilogue: element (m, n) with m = v + 8*khalf, n = row.
    const int ebase = t * 16;
#pragma unroll
    for (int v = 0; v < 8; ++v) {
      const int m = v + 8 * khalf;
      const int e = ebase + m;
      const int s = src_idx[e];
      const int d = dst_idx[e];
      const float w = edge_w[e];
      const float val = (c[v] + h[(size_t)s * HDIM + row]) * w;
      atomicAdd(out + (size_t)d * HDIM + row, val);
    }

    dbuf ^= 1;
  }
}

// ---------------------------------------------------------------------------
// Host-side launch. Inputs (setup_inputs order):
//   0 feat [100000,16] f32      1 review_feat [6400000,64] f32
//   2 edge_w [6400000,1] f32    3 src_idx [6400000] i32
//   4 dst_idx [6400000] i32     5 W_node [16,16] f32
//   6 W_review [16,64] f32
// Output: [100000,16] f32. Workspace: h = 100000*16*4 = 6.4 MB.
// ---------------------------------------------------------------------------
extern "C" void kernel_launch(void* const* d_in, const int* in_sizes, int n_in,
                              void* d_out, int out_size, void* d_ws,
                              size_t ws_size, hipStream_t stream) {
  const float* feat        = (const float*)d_in[0];
  const float* review_feat = (const float*)d_in[1];
  const float* edge_w      = (const float*)d_in[2];
  const int*   src_idx     = (const int*)d_in[3];
  const int*   dst_idx     = (const int*)d_in[4];
  const float* W_node      = (const float*)d_in[5];
  const float* W_review    = (const float*)d_in[6];
  float* out = (float*)d_out;
  float* h   = (float*)d_ws;  // [N_NODES, 16] f32

  const int nout = N_NODES * HDIM;
  zero_out_kernel<<<(nout + 255) / 256, 256, 0, stream>>>(out, nout);

  // 6250 node tiles; 512 blocks * 8 waves = 4096 waves.
  node_h_kernel<<<512, 256, 0, stream>>>(feat, W_node, h);

  // 400000 edge tiles; 4096 blocks * 8 waves = 32768 waves (~12 tiles/wave).
  edge_kernel<<<4096, 256, 0, stream>>>(review_feat, edge_w, src_idx, dst_idx,
                                        W_review, h, out);
}